// DistLoss_31602369364238
// MI455X (gfx1250) — compile-verified
//
#include <hip/hip_runtime.h>
#include <cstdint>

#define W 1024
#define H 1024
#define PADW 1028           // word offset of g_out row inside LDS (16B aligned, bank-shifted by 4)

typedef float v2f __attribute__((ext_vector_type(2)));
typedef float v8f __attribute__((ext_vector_type(8)));

// Explicit LDS (addrspace 3) pointer types: guarantees ds_load_* for the hot loop.
typedef __attribute__((address_space(3))) float lds_f;
typedef __attribute__((address_space(3))) v2f   lds_v2f;

// ---------------------------------------------------------------------------
// Pass 1: per-column vertical distance scans (forward + backward).
// One thread per column; all global accesses are lane-coalesced.
// Produces g_out = vertical-sq-dist for mask=t (sites = bg pixels)
//          g_in  = vertical-sq-dist for mask=~t (sites = fg pixels)
// Matches reference: d = min(up, down); g = d>=H ? 1e12 : d*d.
// ---------------------------------------------------------------------------
__global__ void edt_vertical_kernel(const int* __restrict__ target,
                                    uint32_t* __restrict__ ups,
                                    float* __restrict__ g_out,
                                    float* __restrict__ g_in) {
    const int c = blockIdx.x * blockDim.x + threadIdx.x;

    int last_bg = -H;   // cummax(where(bg, idx, -H)) tracker
    int last_fg = -H;
    for (int r = 0; r < H; ++r) {
        const int fg = (target[r * W + c] != 0) ? 1 : 0;
        if (fg) last_fg = r; else last_bg = r;          // inclusive of current row
        const uint32_t uo = (uint32_t)(r - last_bg);    // <= 2047, fits 12 bits
        const uint32_t ui = (uint32_t)(r - last_fg);
        ups[r * W + c] = uo | (ui << 12) | ((uint32_t)fg << 24);
    }

    int next_bg = 2 * H;  // cummin(where(bg, idx, 2H), reverse) tracker
    int next_fg = 2 * H;
    for (int r = H - 1; r >= 0; --r) {
        const uint32_t u = ups[r * W + c];
        const int fg = (int)((u >> 24) & 1u);
        if (fg) next_fg = r; else next_bg = r;          // inclusive of current row
        const int up_o = (int)(u & 0xFFFu);
        const int up_i = (int)((u >> 12) & 0xFFFu);
        const int d_o = min(up_o, next_bg - r);
        const int d_i = min(up_i, next_fg - r);
        g_out[r * W + c] = (d_o >= H) ? 1e12f : (float)(d_o * d_o);
        g_in [r * W + c] = (d_i >= H) ? 1e12f : (float)(d_i * d_i);
    }
}

// ---------------------------------------------------------------------------
// Pass 2: per-row lower envelope  env[j] = min_k g_sel[k] + (j-k)^2
// fused with the final product/sign and a deterministic row reduction.
// g rows staged into LDS via gfx1250 async LDS loads; the two arrays are 4
// words apart so the per-lane base selection stays bank-disjoint (broadcast-
// friendly ds_load_b64 in the inner loop).
// Contribution per pixel:  fg: x*(1 - sqrt(env_out))   bg: x*sqrt(env_in)
// ---------------------------------------------------------------------------
__global__ void envelope_reduce_kernel(const float* __restrict__ input,
                                       const int* __restrict__ target,
                                       const float* __restrict__ g_out,
                                       const float* __restrict__ g_in,
                                       float* __restrict__ rowp) {
    __shared__ float sh[PADW + W];   // [0..1023] = g_in row, [PADW..PADW+1023] = g_out row
    __shared__ float wsum[8];        // one partial per wave

    const int r = blockIdx.x;
    const int t = threadIdx.x;

    const unsigned lds_base = (unsigned)(uintptr_t)(void*)sh;   // 32-bit LDS offset

    // Async-copy both g rows (4KB each) into LDS: 16 bytes per thread per row.
    {
        const unsigned lds_in  = lds_base + 16u * (unsigned)t;
        const unsigned lds_out = lds_base + 4u * PADW + 16u * (unsigned)t;
        const uint64_t ga_in  = (uint64_t)(uintptr_t)(g_in  + (size_t)r * W) + 16u * (unsigned)t;
        const uint64_t ga_out = (uint64_t)(uintptr_t)(g_out + (size_t)r * W) + 16u * (unsigned)t;
        asm volatile("global_load_async_to_lds_b128 %0, %1, off"
                     :: "v"(lds_in), "v"(ga_in) : "memory");
        asm volatile("global_load_async_to_lds_b128 %0, %1, off"
                     :: "v"(lds_out), "v"(ga_out) : "memory");
        asm volatile("s_wait_asynccnt 0" ::: "memory");
    }
    __syncthreads();

    // Each thread owns 4 output pixels: j = t + 256*s
    const lds_f* const base = (const lds_f*)lds_base;
    const lds_f* p[4];
    float fmul[4], best[4], dj[4];
    float acc = 0.0f;
    #pragma unroll
    for (int s = 0; s < 4; ++s) {
        const int j = t + 256 * s;
        const float x = input[(size_t)r * W + j];
        const bool fg = (target[(size_t)r * W + j] != 0);
        p[s]    = fg ? (base + PADW) : base;   // pick envelope source once per pixel
        fmul[s] = fg ? -x : x;                 // sign of sqrt term
        acc    += fg ? x : 0.0f;               // the "+ input*tf" term
        best[s] = 1e30f;
        dj[s]   = (float)j;                    // incremental (j - k)
    }

    // Brute-force exact envelope; 2 k's per ds_load_b64, broadcast addresses.
    #pragma unroll 4
    for (int k = 0; k < W; k += 2) {
        #pragma unroll
        for (int s = 0; s < 4; ++s) {
            const v2f v = *(const lds_v2f*)(p[s] + k);
            const float d0 = dj[s];
            const float d1 = dj[s] - 1.0f;
            best[s] = fminf(best[s], fmaf(d0, d0, v[0]));
            best[s] = fminf(best[s], fmaf(d1, d1, v[1]));
            dj[s] -= 2.0f;
        }
    }

    #pragma unroll
    for (int s = 0; s < 4; ++s)
        acc += fmul[s] * sqrtf(best[s]);

    // Deterministic reduction: wave shuffle reduce, then fixed-order combine.
    #pragma unroll
    for (int m = 16; m >= 1; m >>= 1)
        acc += __shfl_xor(acc, m, 32);
    if ((t & 31) == 0) wsum[t >> 5] = acc;
    __syncthreads();
    if (t == 0) {
        float s = 0.0f;
        #pragma unroll
        for (int i = 0; i < 8; ++i) s += wsum[i];
        rowp[r] = s;
    }
}

// ---------------------------------------------------------------------------
// Pass 3: sum 1024 row partials with chained V_WMMA_F32_16X16X4_F32.
// A = all-ones => D[i][j] = sum_k B[k][j] (+C). After 16 accumulating WMMAs,
// summing any single C register over all 32 lanes yields 2x the grand total
// (each C VGPR holds two full rows of the 16x16 result).
// ---------------------------------------------------------------------------
__global__ void final_reduce_wmma_kernel(const float* __restrict__ rowp,
                                         float* __restrict__ out) {
    const int lane = threadIdx.x;
    v8f c = {};
    v2f a;
    a[0] = 1.0f; a[1] = 1.0f;

    #pragma unroll
    for (int i = 0; i < 16; ++i) {
        v2f b;
        b[0] = rowp[i * 64 + lane];
        b[1] = rowp[i * 64 + 32 + lane];
        c = __builtin_amdgcn_wmma_f32_16x16x4_f32(
                /*neg_a=*/false, a, /*neg_b=*/false, b,
                /*c_mod=*/(short)0, c, /*reuse_a=*/false, /*reuse_b=*/false);
    }

    float v = c[0];
    #pragma unroll
    for (int m = 16; m >= 1; m >>= 1)
        v += __shfl_xor(v, m, 32);

    if (lane == 0)
        out[0] = (v * 0.5f) * (1.0f / 1048576.0f);   // mean over 1024*1024
}

// ---------------------------------------------------------------------------
extern "C" void kernel_launch(void* const* d_in, const int* in_sizes, int n_in,
                              void* d_out, int out_size, void* d_ws, size_t ws_size,
                              hipStream_t stream) {
    const float* input  = (const float*)d_in[0];
    const int*   target = (const int*)d_in[1];

    char* ws = (char*)d_ws;
    float*    g_out = (float*)(ws);                         // 4 MB
    float*    g_in  = (float*)(ws + (size_t)(4  << 20));    // 4 MB
    uint32_t* ups   = (uint32_t*)(ws + (size_t)(8 << 20));  // 4 MB
    float*    rowp  = (float*)(ws + (size_t)(12 << 20));    // 4 KB

    edt_vertical_kernel<<<dim3(W / 256), dim3(256), 0, stream>>>(target, ups, g_out, g_in);
    envelope_reduce_kernel<<<dim3(H), dim3(256), 0, stream>>>(input, target, g_out, g_in, rowp);
    final_reduce_wmma_kernel<<<dim3(1), dim3(32), 0, stream>>>(rowp, (float*)d_out);
}